// GraphAttention_30073361006900
// MI455X (gfx1250) — compile-verified
//
#include <hip/hip_runtime.h>

#define B_ 8
#define N_ 1024
#define D_ 256
#define H_ 8
#define U_ 64
#define HU_ 512
#define EROW 1028  // LDS row stride in floats; 1028 % 64 == 4 -> conflict-free b64

typedef __attribute__((ext_vector_type(2))) float v2f;
typedef __attribute__((ext_vector_type(8))) float v8f;

__device__ __forceinline__ v8f wmma_f32(v2f a, v2f b, v8f c) {
  // D = A(16x4) * B(4x16) + C(16x16), fp32
  return __builtin_amdgcn_wmma_f32_16x16x4_f32(
      /*neg_a=*/false, a, /*neg_b=*/false, b,
      /*c_mod=*/(short)0, c, /*reuse_a=*/false, /*reuse_b=*/false);
}

// ---------------------------------------------------------------------------
// Phase 1: features = X @ Wk ; residual = X @ Wr    (M=8192, N=512, K=256)
// One wave per 16x16 output tile; A-tile of X shared by both WMMA chains.
// ---------------------------------------------------------------------------
__global__ void gat_proj(const float* __restrict__ X,
                         const float* __restrict__ Wk,
                         const float* __restrict__ Wr,
                         float* __restrict__ feat,
                         float* __restrict__ resid) {
  const int wave = threadIdx.x >> 5;
  const int lane = threadIdx.x & 31;
  const int tile = blockIdx.x * (blockDim.x >> 5) + wave;
  const int NT = HU_ / 16;                 // 32 col tiles
  const int mt = tile / NT, nt = tile % NT;
  const int row0 = mt * 16, col0 = nt * 16;
  const int r = lane & 15, hf = lane >> 4;

  v8f accF = {};
  v8f accR = {};
  const float* xrow = X + (size_t)(row0 + r) * D_;
  for (int kb = 0; kb < D_; kb += 4) {
    const int k0 = kb + 2 * hf;            // this lane-half's K pair
    v2f a;
    a[0] = xrow[k0];
    a[1] = xrow[k0 + 1];
    v2f bf, br;
    bf[0] = Wk[(size_t)k0 * HU_ + col0 + r];
    bf[1] = Wk[(size_t)(k0 + 1) * HU_ + col0 + r];
    br[0] = Wr[(size_t)k0 * HU_ + col0 + r];
    br[1] = Wr[(size_t)(k0 + 1) * HU_ + col0 + r];
    accF = wmma_f32(a, bf, accF);
    accR = wmma_f32(a, br, accR);
  }
#pragma unroll
  for (int i = 0; i < 8; ++i) {
    const int row = row0 + i + 8 * hf;     // C layout: VGPR i -> row i + 8*half
    feat[(size_t)row * HU_ + col0 + r] = accF[i];
    resid[(size_t)row * HU_ + col0 + r] = accR[i];
  }
}

// ---------------------------------------------------------------------------
// Phase 1b: selfT[b,h,n] = sum_u f[b,n,h,u]*aks[h,u]; neighT likewise.
// One wave per (b,n,h); coalesced 64-float read, wave32 shuffle reduction.
// ---------------------------------------------------------------------------
__global__ void gat_attn_logits(const float* __restrict__ feat,
                                const float* __restrict__ aks,
                                const float* __restrict__ akn,
                                float* __restrict__ selfT,
                                float* __restrict__ neighT) {
  const int wave = threadIdx.x >> 5;
  const int lane = threadIdx.x & 31;
  const int idx = blockIdx.x * (blockDim.x >> 5) + wave;  // (b*N + n)*H + h
  const int h = idx % H_;
  const int bn = idx / H_;
  const float* fp = feat + (size_t)bn * HU_ + h * U_;
  const float f0 = fp[lane], f1 = fp[lane + 32];
  float s0 = f0 * aks[h * U_ + lane] + f1 * aks[h * U_ + lane + 32];
  float s1 = f0 * akn[h * U_ + lane] + f1 * akn[h * U_ + lane + 32];
#pragma unroll
  for (int off = 16; off > 0; off >>= 1) {
    s0 += __shfl_xor(s0, off, 32);
    s1 += __shfl_xor(s1, off, 32);
  }
  if (lane == 0) {
    const int b = bn / N_, n = bn % N_;
    selfT[((size_t)b * H_ + h) * N_ + n] = s0;
    neighT[((size_t)b * H_ + h) * N_ + n] = s1;
  }
}

// ---------------------------------------------------------------------------
// Phase 2: one block (2 waves) per (b,h,16-row tile).
// Pass 1 (k-range split across the 2 waves): e = exp(masked leaky logit),
// stored to LDS directly in WMMA-A layout order; row sums merged via LDS.
// Logits are bounded (masked -> exp == 0), so no max-subtraction is needed
// and exp is evaluated exactly once per (row,k).
// Pass 2: each wave accumulates P @ F for its half of the U=64 columns with
// fp32 WMMA, reading probabilities back via conflict-free ds_load_b64.
// ---------------------------------------------------------------------------
__global__ void __launch_bounds__(64) gat_attend(
    const float* __restrict__ A,
    const float* __restrict__ feat,
    const float* __restrict__ resid,
    const float* __restrict__ selfT,
    const float* __restrict__ neighT,
    const float* __restrict__ bias,
    float* __restrict__ out) {
  __shared__ float lds_e[16 * EROW];   // exp(logit) tile, padded rows
  __shared__ float lds_s[2][16];       // per-wave partial row sums

  const int wave = threadIdx.x >> 5;   // 0..1
  const int lane = threadIdx.x & 31;
  const int tile = blockIdx.x;         // (b*H + h)*64 + nt
  const int nt = tile & 63;
  const int bh = tile >> 6;
  const int h = bh % H_;
  const int b = bh / H_;
  const int n0 = nt * 16;
  const int r = lane & 15, hf = lane >> 4;

  const float* Arow = A + ((size_t)b * N_ + n0 + r) * N_;
  const float* ngRow = neighT + (size_t)bh * N_;
  const float aself = selfT[(size_t)bh * N_ + n0 + r];

  // ---- pass 1: single-sweep exp + sum; each wave handles half the k range --
  float s = 0.0f;
  const int kbeg = wave * (N_ / 2);
  const int kend = kbeg + (N_ / 2);
  for (int kb = kbeg; kb < kend; kb += 4) {
    const int k = kb + 2 * hf;                 // WMMA-A layout: K = j + 2*half
    const float2 av = *(const float2*)(Arow + k);
    const float2 ng = *(const float2*)(ngRow + k);
    float l0 = aself + ng.x;
    l0 = (l0 > 0.0f) ? l0 : 0.2f * l0;         // leaky_relu(0.2)
    l0 += -1.0e10f * (1.0f - av.x);            // mask non-edges
    float l1 = aself + ng.y;
    l1 = (l1 > 0.0f) ? l1 : 0.2f * l1;
    l1 += -1.0e10f * (1.0f - av.y);
    const float e0 = __expf(l0);               // masked -> exactly 0
    const float e1 = __expf(l1);
    s += e0 + e1;
    float2 st; st.x = e0; st.y = e1;
    *(float2*)(&lds_e[r * EROW + k]) = st;     // conflict-free ds_store_b64
  }
  s += __shfl_xor(s, 16, 32);                  // merge the two lane-halves
  if (lane < 16) lds_s[wave][r] = s;
  __syncthreads();
  const float rinv = 1.0f / (lds_s[0][r] + lds_s[1][r]);

  // ---- pass 2: P @ F via WMMA; this wave covers u in [u0, u0+32) ----------
  const int u0 = wave * 32;
  v8f accA = {}, accB = {};
  const float* fbase = feat + (size_t)b * N_ * HU_ + h * U_ + u0 + r;
  for (int kb = 0; kb < N_; kb += 4) {
    const int k0 = kb + 2 * hf;
    const float2 e2 = *(const float2*)(&lds_e[r * EROW + k0]);
    v2f p;
    p[0] = e2.x * rinv;
    p[1] = e2.y * rinv;
    const float* f0 = fbase + (size_t)k0 * HU_;
    const float* f1 = f0 + HU_;
    v2f bA, bB;
    bA[0] = f0[0];  bA[1] = f1[0];
    bB[0] = f0[16]; bB[1] = f1[16];
    accA = wmma_f32(p, bA, accA);
    accB = wmma_f32(p, bB, accB);
  }

  // ---- store: relu(acc + residual + bias) ----
#pragma unroll
  for (int i = 0; i < 8; ++i) {
    const int row = n0 + i + 8 * hf;
    const size_t base = ((size_t)b * N_ + row) * HU_ + h * U_ + u0 + r;
    const float v0 = accA[i] + resid[base]      + bias[h * U_ + u0 + r];
    const float v1 = accB[i] + resid[base + 16] + bias[h * U_ + u0 + 16 + r];
    out[base]      = fmaxf(v0, 0.0f);
    out[base + 16] = fmaxf(v1, 0.0f);
  }
}

// ---------------------------------------------------------------------------
extern "C" void kernel_launch(void* const* d_in, const int* in_sizes, int n_in,
                              void* d_out, int out_size, void* d_ws, size_t ws_size,
                              hipStream_t stream) {
  const float* X    = (const float*)d_in[0];
  const float* Adj  = (const float*)d_in[1];
  const float* Wk   = (const float*)d_in[2];
  const float* Wr   = (const float*)d_in[3];
  const float* aks  = (const float*)d_in[4];
  const float* akn  = (const float*)d_in[5];
  const float* bias = (const float*)d_in[6];
  float* out = (float*)d_out;

  float* feat   = (float*)d_ws;                         // 8192*512 f32
  float* resid  = feat  + (size_t)B_ * N_ * HU_;        // 8192*512 f32
  float* selfT  = resid + (size_t)B_ * N_ * HU_;        // 8*8*1024 f32
  float* neighT = selfT + (size_t)B_ * H_ * N_;         // 8*8*1024 f32

  {  // Phase 1: projections. 16384 wave-tiles, 8 waves/block.
    const int tiles = (B_ * N_ / 16) * (HU_ / 16);
    gat_proj<<<dim3(tiles / 8), dim3(256), 0, stream>>>(X, Wk, Wr, feat, resid);
  }
  {  // Phase 1b: attention logits. 65536 waves, 8 waves/block.
    const int items = B_ * N_ * H_;
    gat_attn_logits<<<dim3(items / 8), dim3(256), 0, stream>>>(feat, aks, akn,
                                                               selfT, neighT);
  }
  {  // Phase 2: softmax + aggregation. One 2-wave block per 16-row tile.
    const int tiles = B_ * H_ * (N_ / 16);               // 4096
    gat_attend<<<dim3(tiles), dim3(64), 0, stream>>>(Adj, feat, resid,
                                                     selfT, neighT, bias, out);
  }
}